// RPN_90340342104768
// MI455X (gfx1250) — compile-verified
//
#include <hip/hip_runtime.h>
#include <stdint.h>

#define GTN 32
#define POS_T 0.7f
#define NEG_T 0.3f

typedef float v4f __attribute__((ext_vector_type(4)));   // native vector for NT stores

// ---------------------------------------------------------------------------
// Main kernel (placed first so the disasm snippet shows it).
// One image per blockIdx.y, grid-stride over anchors.
//   out layout (floats): labels [B*N] | matched [B*N*4] | scores [B*N]
// ---------------------------------------------------------------------------
__global__ __launch_bounds__(256) void rpn_match(
    const float4* __restrict__ anchors,   // [N] xyxy
    const float*  __restrict__ gt,        // [B,G,4] xywh
    float*        __restrict__ out,
    unsigned long long* __restrict__ red, // [B*G] packed argmax slots
    int N)
{
    const int b   = blockIdx.y;
    const int tid = threadIdx.x;

    __shared__ float4 s_raw[GTN];                 // raw xywh (async target)
    __shared__ float4 s_box[GTN];                 // xyxy
    __shared__ float  s_area[GTN];                // gt areas
    __shared__ unsigned long long s_red[GTN];     // block-level argmax

    // --- Stage GT boxes for this image into LDS via CDNA5 async-to-LDS ---
    if (tid < GTN) {
        s_red[tid] = 0ULL;
        uint32_t ldsdst = (uint32_t)(uintptr_t)(&s_raw[tid]); // low32 = LDS offset
        uint32_t voff   = (uint32_t)tid * 16u;
        uint64_t sbase  = (uint64_t)(uintptr_t)(gt + (size_t)b * GTN * 4);
        asm volatile(
            "global_load_async_to_lds_b128 %0, %1, %2 offset:0\n\t"
            "s_wait_asynccnt 0"
            :: "v"(ldsdst), "v"(voff), "s"(sbase) : "memory");
    }
    __syncthreads();
    if (tid < GTN) {
        float4 r = s_raw[tid];                    // x, y, w, h
        float4 x = make_float4(r.x, r.y, r.x + r.z, r.y + r.w);
        s_box[tid]  = x;
        s_area[tid] = (x.z - x.x) * (x.w - x.y);  // same expr as reference
    }
    __syncthreads();

    // Per-thread running per-GT argmax (64 VGPRs after full unroll).
    float    runMax[GTN];
    uint32_t runIdx[GTN];
    #pragma unroll
    for (int g = 0; g < GTN; ++g) { runMax[g] = 0.0f; runIdx[g] = 0u; }

    const int stride = gridDim.x * blockDim.x;

    float* lab_out = out;                                  // + b*N + i
    float* mat_out = out + (size_t)2 * N;                  // + (b*N + i)*4
    float* sc_out  = out + (size_t)10 * N;                 // + b*N + i

    for (int i = blockIdx.x * blockDim.x + tid; i < N; i += stride) {
        __builtin_prefetch(&anchors[i + stride], 0, 1);    // global_prefetch
        float4 a = anchors[i];
        float areaA = (a.z - a.x) * (a.w - a.y);

        float best  = -1.0f;
        int   bestG = 0;
        #pragma unroll
        for (int g = 0; g < GTN; ++g) {
            float4 gb = s_box[g];                  // hoisted to VGPRs by LICM
            float ix1 = fmaxf(a.x, gb.x);
            float iy1 = fmaxf(a.y, gb.y);
            float ix2 = fminf(a.z, gb.z);
            float iy2 = fminf(a.w, gb.w);
            float dw  = fmaxf(ix2 - ix1, 0.0f);
            float dh  = fmaxf(iy2 - iy1, 0.0f);
            float inter = dw * dh;
            float uni   = (areaA + s_area[g]) - inter;
            float iou   = inter * __builtin_amdgcn_rcpf(fmaxf(uni, 1e-9f));

            bool bA = iou > best;                  // first-max tie-break over g
            bestG   = bA ? g : bestG;
            best    = fmaxf(best, iou);

            bool bG   = iou > runMax[g];           // first-max over anchors
            runIdx[g] = bG ? (uint32_t)i : runIdx[g];
            runMax[g] = fmaxf(runMax[g], iou);
        }

        float lab = (best > POS_T) ? 1.0f : ((best < NEG_T) ? 0.0f : -1.0f);
        size_t o = (size_t)b * N + i;
        // Streaming outputs: non-temporal stores (th:TH_STORE_NT)
        __builtin_nontemporal_store(lab, &lab_out[o]);
        __builtin_nontemporal_store(best, &sc_out[o]);
        float4 sb = s_box[bestG];
        v4f m = { sb.x, sb.y, sb.z, sb.w };
        __builtin_nontemporal_store(m, (v4f*)(mat_out + o * 4));
    }

    // --- Reduce per-GT argmax: regs -> LDS (ds_max_u64) -> global atomic ---
    #pragma unroll
    for (int g = 0; g < GTN; ++g) {
        unsigned long long packed =
            ((unsigned long long)__float_as_uint(runMax[g]) << 32) |
            (unsigned long long)(0xFFFFFFFFu - runIdx[g]);   // smaller idx wins ties
        atomicMax(&s_red[g], packed);
    }
    __syncthreads();
    if (tid < GTN) {
        atomicMax(&red[b * GTN + tid], s_red[tid]);
    }
}

// ---------------------------------------------------------------------------
// Kernel 0: zero the 64 packed (score,idx) argmax slots in workspace.
// ---------------------------------------------------------------------------
__global__ void rpn_init_ws(unsigned long long* __restrict__ red) {
    int i = threadIdx.x;
    if (i < 2 * GTN) red[i] = 0ULL;
}

// ---------------------------------------------------------------------------
// Kernel 2: force-match best anchor per GT -> label 1.0
// ---------------------------------------------------------------------------
__global__ void rpn_fixup(const unsigned long long* __restrict__ red,
                          float* __restrict__ out, int N)
{
    int i = threadIdx.x;
    if (i >= 2 * GTN) return;
    int b = i >> 5;
    unsigned long long p = red[i];
    uint32_t idx = 0xFFFFFFFFu - (uint32_t)(p & 0xFFFFFFFFull);
    if (idx < (uint32_t)N) out[(size_t)b * N + idx] = 1.0f;
}

// ---------------------------------------------------------------------------
extern "C" void kernel_launch(void* const* d_in, const int* in_sizes, int n_in,
                              void* d_out, int out_size, void* d_ws, size_t ws_size,
                              hipStream_t stream) {
    const float4* anchors = (const float4*)d_in[0];   // [N,4] xyxy
    const float*  gt      = (const float*)d_in[1];    // [2,32,4] xywh
    float* out = (float*)d_out;
    unsigned long long* red = (unsigned long long*)d_ws;  // 64 * 8 bytes
    const int N = in_sizes[0] / 4;

    rpn_init_ws<<<1, 64, 0, stream>>>(red);

    dim3 grid(512, 2);   // ~131K threads/image -> ~10 anchors/thread; 32K atomics total
    rpn_match<<<grid, 256, 0, stream>>>(anchors, gt, out, red, N);

    rpn_fixup<<<1, 64, 0, stream>>>(red, out, N);
}